// FilterShortRange_9569187135587
// MI455X (gfx1250) — compile-verified
//
#include <hip/hip_runtime.h>
#include <stdint.h>

#define THREADS 256
#define IPT 8
#define CHUNK (THREADS * IPT)   // 2048 edges per block

typedef uint32_t u32;
typedef uint32_t u32x4 __attribute__((ext_vector_type(4)));
typedef uint32_t u32x8 __attribute__((ext_vector_type(8)));

// Shared mask predicate: MUST be identical in count and compact kernels so the
// per-block popcounts and per-element ranks agree bit-exactly.
__device__ __forceinline__ bool edge_in(float x, float y, float z) {
    float d = fmaf(x, x, fmaf(y, y, z * z));
    return sqrtf(d) <= 1.6f;
}

// ---------------------------------------------------------------------------
// Kernel 1: pass-through copies + per-block short-range count.
// ---------------------------------------------------------------------------
__global__ void __launch_bounds__(THREADS)
fsr_pass_count(const float* __restrict__ Rij, const int* __restrict__ idxI,
               const int* __restrict__ idxJ, float* __restrict__ outRij,
               float* __restrict__ outI, float* __restrict__ outJ,
               u32* __restrict__ counts, int E)
{
    const int t = threadIdx.x;
    const int b = blockIdx.x;
    const int base = b * CHUNK;
    const int e0 = base + t * IPT;
    u32 cnt = 0;

    if (e0 + IPT <= E) {
        // Fully vectorized path: 96B Rij copy (6x float4), 32B idx loads (2x int4).
        float4 v[6];
        const float4* src = (const float4*)(Rij + (size_t)3 * e0);
        float4* dst = (float4*)(outRij + (size_t)3 * e0);
#pragma unroll
        for (int k = 0; k < 6; ++k) v[k] = src[k];
#pragma unroll
        for (int k = 0; k < 6; ++k) dst[k] = v[k];
        const float* f = (const float*)v;

        const int4* pi = (const int4*)(idxI + e0);
        const int4* pj = (const int4*)(idxJ + e0);
        int4 a0 = pi[0], a1 = pi[1];
        int4 c0 = pj[0], c1 = pj[1];
        float4* oi = (float4*)(outI + e0);
        float4* oj = (float4*)(outJ + e0);
        oi[0] = make_float4((float)a0.x, (float)a0.y, (float)a0.z, (float)a0.w);
        oi[1] = make_float4((float)a1.x, (float)a1.y, (float)a1.z, (float)a1.w);
        oj[0] = make_float4((float)c0.x, (float)c0.y, (float)c0.z, (float)c0.w);
        oj[1] = make_float4((float)c1.x, (float)c1.y, (float)c1.z, (float)c1.w);

#pragma unroll
        for (int k = 0; k < IPT; ++k)
            cnt += edge_in(f[3 * k], f[3 * k + 1], f[3 * k + 2]) ? 1u : 0u;
    } else {
        for (int k = 0; k < IPT; ++k) {
            int e = e0 + k;
            if (e < E) {
                size_t e3 = (size_t)3 * e;
                float x = Rij[e3], y = Rij[e3 + 1], z = Rij[e3 + 2];
                outRij[e3] = x; outRij[e3 + 1] = y; outRij[e3 + 2] = z;
                outI[e] = (float)idxI[e];
                outJ[e] = (float)idxJ[e];
                cnt += edge_in(x, y, z) ? 1u : 0u;
            }
        }
    }

    __shared__ u32 sc[THREADS];
    sc[t] = cnt;
    __syncthreads();
#pragma unroll
    for (int s2 = THREADS / 2; s2 > 0; s2 >>= 1) {
        if (t < s2) sc[t] += sc[t + s2];
        __syncthreads();
    }
    if (t == 0) counts[b] = sc[0];
}

// ---------------------------------------------------------------------------
// Kernel 2: exclusive scan of per-block counts (single block, 1024 threads).
// ---------------------------------------------------------------------------
__global__ void __launch_bounds__(1024)
fsr_scan(const u32* __restrict__ counts, u32* __restrict__ offsets,
         u32* __restrict__ nsr, float* __restrict__ outN, int nblocks)
{
    __shared__ u32 s[1024];
    const int t = threadIdx.x;
    const int items = (nblocks + 1023) >> 10;
    const int lo = t * items;
    const int hi = min(lo + items, nblocks);

    u32 sum = 0;
    for (int i = lo; i < hi; ++i) sum += counts[i];
    s[t] = sum;
    __syncthreads();

    // Hillis-Steele inclusive scan over 1024 partials.
    for (int off = 1; off < 1024; off <<= 1) {
        u32 v = s[t];
        if (t >= off) v += s[t - off];
        __syncthreads();
        s[t] = v;
        __syncthreads();
    }

    u32 run = s[t] - sum;  // exclusive base for this thread's range
    for (int i = lo; i < hi; ++i) {
        offsets[i] = run;
        run += counts[i];
    }
    if (t == 1023) {
        nsr[0] = s[1023];
        outN[0] = (float)s[1023];
    }
}

// ---------------------------------------------------------------------------
// Kernel 3: stable scatter. Rij tile staged into LDS by the Tensor Data Mover;
// the per-thread idx loads (LOADcnt) are issued between the TDM issue and the
// TENSORcnt drain so VMEM and TDM traffic overlap.
// ---------------------------------------------------------------------------
__global__ void __launch_bounds__(THREADS)
fsr_compact(const float* __restrict__ Rij, const int* __restrict__ idxI,
            const int* __restrict__ idxJ, const u32* __restrict__ offsets,
            float* __restrict__ outRijSr, float* __restrict__ outISr,
            float* __restrict__ outJSr, int E)
{
    __shared__ float tile[CHUNK * 3];   // 24 KB
    __shared__ u32 sc[THREADS];
    const int t = threadIdx.x;
    const int b = blockIdx.x;
    const int base = b * CHUNK;
    const int nvalid = min(CHUNK, E - base);

    // --- Issue TDM load of the block's Rij tile (3*nvalid floats, 1-row tile,
    // data_size=4B) into LDS. TDM ignores EXEC; only wave 0 reaches this block.
    if (t == 0) {
        const u32 nflt = (u32)nvalid * 3u;                       // <= 6144, fits 16b tile_dim0
        const u32 lds  = (u32)(uintptr_t)&tile[0];               // low 32b = LDS byte address
        const uint64_t ga = (uint64_t)(uintptr_t)(Rij + (size_t)3 * base);
        u32x4 g0;
        g0[0] = 1u;                                              // count=1 (valid), user mode
        g0[1] = lds;                                             // lds_addr
        g0[2] = (u32)ga;                                         // global_addr[31:0]
        g0[3] = (u32)((ga >> 32) & 0x1FFFFFFu) | (2u << 30);     // addr[56:32] | type=2
        u32x8 g1;
        g1[0] = 2u << 16;                                        // data_size=4B; no multicast
        g1[1] = (nflt & 0xFFFFu) << 16;                          // tensor_dim0[15:0]
        g1[2] = (nflt >> 16) | (1u << 16);                       // tensor_dim0[31:16] | tensor_dim1=1
        g1[3] = nflt << 16;                                      // tile_dim0 = 3*nvalid
        g1[4] = 0u;                                              // tile_dim1/2 unused
        g1[5] = nflt;                                            // tensor_dim0_stride
        g1[6] = 0u;
        g1[7] = 0u;
        asm volatile("tensor_load_to_lds %0, %1" :: "s"(g0), "s"(g1) : "memory");
    }

    // --- Overlap: independent global loads while the TDM transfer is in flight.
    const int e0 = base + t * IPT;
    int ia[IPT], ja[IPT];
    if (e0 + IPT <= E) {
        const int4* pi = (const int4*)(idxI + e0);
        const int4* pj = (const int4*)(idxJ + e0);
        int4 a0 = pi[0], a1 = pi[1], c0 = pj[0], c1 = pj[1];
        ia[0] = a0.x; ia[1] = a0.y; ia[2] = a0.z; ia[3] = a0.w;
        ia[4] = a1.x; ia[5] = a1.y; ia[6] = a1.z; ia[7] = a1.w;
        ja[0] = c0.x; ja[1] = c0.y; ja[2] = c0.z; ja[3] = c0.w;
        ja[4] = c1.x; ja[5] = c1.y; ja[6] = c1.z; ja[7] = c1.w;
    } else {
#pragma unroll
        for (int k = 0; k < IPT; ++k) {
            int e = e0 + k;
            ia[k] = (e < E) ? idxI[e] : -1;
            ja[k] = (e < E) ? idxJ[e] : -1;
        }
    }
    const u32 blockOff = offsets[b];

    // --- Drain TENSORcnt (issuing wave) and make the tile visible block-wide.
    if (t == 0) {
        __builtin_amdgcn_s_wait_tensorcnt(0);
    }
    __syncthreads();

    float xs[IPT], ys[IPT], zs[IPT];
    u32 cnt = 0, m = 0;
#pragma unroll
    for (int k = 0; k < IPT; ++k) {
        int e = e0 + k;
        bool p = false;
        xs[k] = 0.f; ys[k] = 0.f; zs[k] = 0.f;
        if (e < E) {
            int l = (t * IPT + k) * 3;
            xs[k] = tile[l]; ys[k] = tile[l + 1]; zs[k] = tile[l + 2];
            p = edge_in(xs[k], ys[k], zs[k]);
        }
        m |= (p ? 1u : 0u) << k;
        cnt += p ? 1u : 0u;
    }

    // Block-wide inclusive scan of per-thread counts -> stable exclusive rank.
    sc[t] = cnt;
    __syncthreads();
#pragma unroll
    for (int off = 1; off < THREADS; off <<= 1) {
        u32 v = sc[t];
        if (t >= off) v += sc[t - off];
        __syncthreads();
        sc[t] = v;
        __syncthreads();
    }
    u32 pos = blockOff + (sc[t] - cnt);

#pragma unroll
    for (int k = 0; k < IPT; ++k) {
        if (m & (1u << k)) {
            size_t p3 = (size_t)pos * 3;
            outRijSr[p3] = xs[k];
            outRijSr[p3 + 1] = ys[k];
            outRijSr[p3 + 2] = zs[k];
            outISr[pos] = (float)ia[k];
            outJSr[pos] = (float)ja[k];
            ++pos;
        }
    }
}

// ---------------------------------------------------------------------------
// Kernel 4: padding fill for positions >= n_sr (zeros / -1).
// ---------------------------------------------------------------------------
__global__ void __launch_bounds__(THREADS)
fsr_tail(float* __restrict__ outRijSr, float* __restrict__ outISr,
         float* __restrict__ outJSr, const u32* __restrict__ nsr, int E3, int E)
{
    const int f = blockIdx.x * THREADS + threadIdx.x;
    const u32 n = nsr[0];
    if (f < E3 && (u32)f >= 3u * n) outRijSr[f] = 0.0f;
    if (f < E && (u32)f >= n) {
        outISr[f] = -1.0f;
        outJSr[f] = -1.0f;
    }
}

// ---------------------------------------------------------------------------
extern "C" void kernel_launch(void* const* d_in, const int* in_sizes, int n_in,
                              void* d_out, int out_size, void* d_ws, size_t ws_size,
                              hipStream_t stream)
{
    (void)n_in; (void)out_size; (void)ws_size;
    const float* Rij = (const float*)d_in[0];
    const int* idxI = (const int*)d_in[1];
    const int* idxJ = (const int*)d_in[2];
    const int E = in_sizes[1];          // idx_i element count

    float* out = (float*)d_out;
    float* outRij   = out;
    float* outI     = out + (size_t)3 * E;
    float* outJ     = out + (size_t)4 * E;
    float* outRijSr = out + (size_t)5 * E;
    float* outISr   = out + (size_t)8 * E;
    float* outJSr   = out + (size_t)9 * E;
    float* outN     = out + (size_t)10 * E;

    const int nblocks = (E + CHUNK - 1) / CHUNK;
    u32* counts  = (u32*)d_ws;
    u32* offsets = counts + nblocks;
    u32* nsr     = offsets + nblocks;

    fsr_pass_count<<<nblocks, THREADS, 0, stream>>>(Rij, idxI, idxJ,
                                                    outRij, outI, outJ, counts, E);
    fsr_scan<<<1, 1024, 0, stream>>>(counts, offsets, nsr, outN, nblocks);
    fsr_compact<<<nblocks, THREADS, 0, stream>>>(Rij, idxI, idxJ, offsets,
                                                 outRijSr, outISr, outJSr, E);
    const int E3 = 3 * E;
    fsr_tail<<<(E3 + THREADS - 1) / THREADS, THREADS, 0, stream>>>(outRijSr, outISr,
                                                                   outJSr, nsr, E3, E);
}